// MonotonicOracle_69750268887093
// MI455X (gfx1250) — compile-verified
//
#include <hip/hip_runtime.h>
#include <hip/hip_bf16.h>
#include <math.h>

// MonotonicOracle for MI455X (gfx1250, wave32).
// N=8192 nodes, H=64, B=8, E=262144 edges.
//
// Strategy: never materialize the 256MB dense adj. Use an 8MB bitmask
// (exactly reproduces jnp .set(1.0) dedup + the +eye diagonal doubling),
// do the one real GEMM (h @ W_g^T, 65536x64x64) with V_WMMA_F32_16X16X4_F32
// at full f32 precision, and fuse the SpMM with every downstream reduction
// so z (16MB) is never stored.

#define NN   8192
#define HH   64
#define BB   8
#define ROWW 256           // 8192 bits / 32 = 256 u32 words per adjacency row

typedef float v2f __attribute__((ext_vector_type(2)));
typedef float v8f __attribute__((ext_vector_type(8)));

// ---------------------------------------------------------------- edge scatter
__global__ void k_scatter(const int* __restrict__ ei, unsigned* __restrict__ bm, int E) {
    int e = blockIdx.x * blockDim.x + threadIdx.x;
    if (e >= E) return;
    int r = ei[e];
    int c = ei[E + e];
    atomicOr(&bm[(size_t)r * ROWW + (c >> 5)], 1u << (c & 31));
}

// -------------------------------------------------- degree -> dis = deg^-1/2
// one wave per node; coalesced popcount scan of the 256-word row.
__global__ void k_degree(const unsigned* __restrict__ bm, float* __restrict__ dis) {
    int wave = (blockIdx.x * blockDim.x + threadIdx.x) >> 5;
    int lane = threadIdx.x & 31;
    if (wave >= NN) return;
    const unsigned* row = bm + (size_t)wave * ROWW;
    int cnt = 0;
    for (int w = lane; w < ROWW; w += 32) cnt += __builtin_popcount(row[w]);
    for (int o = 16; o; o >>= 1) cnt += __shfl_down(cnt, o);
    if (lane == 0) dis[wave] = 1.0f / sqrtf((float)(cnt + 1)); // +1 from eye; self-edge bit already counted
}

// ----------------------------------- h = node_emb + x@W_sp^T + b_sp ; h@W_g^T
// 256 threads = 8 waves, 128 rows/block, each wave does a 16x64 tile
// via 64x V_WMMA_F32_16X16X4_F32 from LDS fragments.
__global__ __launch_bounds__(256) void k_hproj(
    const float* __restrict__ x, const float* __restrict__ node_emb,
    const float* __restrict__ W_sp, const float* __restrict__ b_sp,
    const float* __restrict__ W_g, float* __restrict__ h_proj)
{
    __shared__ float sW[64 * 66];   // W_g row-major, stride 66 (bank-conflict pad)
    __shared__ float sH[128 * 66];  // staged h tile
    __shared__ float rx0[128], rx1[128];
    __shared__ float c0[64], c1[64], cb[64];

    int t  = threadIdx.x;
    int g0 = blockIdx.x * 128;

    if (t < 64) { c0[t] = W_sp[t * 2]; c1[t] = W_sp[t * 2 + 1]; cb[t] = b_sp[t]; }
    if (t < 128) {
        int g = g0 + t;
        rx0[t] = x[(size_t)g * 2];
        rx1[t] = x[(size_t)g * 2 + 1];
    }
    for (int i = t; i < 64 * 64; i += 256)
        sW[(i >> 6) * 66 + (i & 63)] = W_g[i];
    __syncthreads();

    for (int i = t; i < 128 * 64; i += 256) {
        int r = i >> 6, c = i & 63;
        int g = g0 + r;
        int n = g & (NN - 1);
        sH[r * 66 + c] = node_emb[(size_t)n * HH + c] + rx0[r] * c0[c] + rx1[r] * c1[c] + cb[c];
    }
    __syncthreads();

    int w = t >> 5, lane = t & 31;
    int lhalf = lane >> 4, lm = lane & 15;
    int wr0 = w * 16;

    v8f acc[4] = {};
#pragma unroll
    for (int kk = 0; kk < 16; ++kk) {
        // A 16x4 f32: lanes 0-15 -> (K=4kk,4kk+1), lanes 16-31 -> (K=4kk+2,4kk+3)
        v2f a = *(const v2f*)&sH[(wr0 + lm) * 66 + kk * 4 + lhalf * 2];
#pragma unroll
        for (int jt = 0; jt < 4; ++jt) {
            // B 4x16: B[k][j] = W_g[j][k]
            v2f bf = *(const v2f*)&sW[(jt * 16 + lm) * 66 + kk * 4 + lhalf * 2];
            acc[jt] = __builtin_amdgcn_wmma_f32_16x16x4_f32(
                false, a, false, bf, (short)0, acc[jt], false, false);
        }
    }

    // D layout: vgpr v, lanes 0-15 -> M=v, lanes 16-31 -> M=v+8, col = lane%16
    size_t grow = (size_t)(g0 + wr0 + lhalf * 8);
#pragma unroll
    for (int jt = 0; jt < 4; ++jt)
#pragma unroll
        for (int v = 0; v < 8; ++v)
            h_proj[(grow + v) * HH + jt * 16 + lm] = acc[jt][v];
}

// ---------------- fused SpMM z = relu(D^-1/2 (A+I) D^-1/2 hproj + b_g) + sinks
// 32 nodes per block; thread t owns (b0=t/64, h=t%64) and (b0+4, h).
__global__ __launch_bounds__(256) void k_spmm(
    const unsigned* __restrict__ bm, const float* __restrict__ dis,
    const float* __restrict__ h_proj, const float* __restrict__ x,
    const float* __restrict__ b_g, const float* __restrict__ msk,
    const int* __restrict__ target,
    float* __restrict__ wz_g, float* __restrict__ sp_g,
    float* __restrict__ nl_g, float* __restrict__ z_target)
{
    __shared__ unsigned row[ROWW];
    __shared__ float spw[16];
    __shared__ float bg[64];

    int t = threadIdx.x;
    int b0 = t >> 6, h = t & 63, b1 = b0 + 4;
    int wv = t >> 5, lane = t & 31;
    if (t < 64) bg[t] = b_g[t];
    int tg0 = target[b0];
    int tg1 = target[b1];

    float wz0 = 0.f, wz1 = 0.f, spaccR = 0.f, nlaccR = 0.f;

    for (int ni = 0; ni < 32; ++ni) {
        int n = blockIdx.x * 32 + ni;
        __syncthreads();                       // protect row/spw reuse
        row[t] = bm[(size_t)n * ROWW + t];
        __syncthreads();

        float dn = dis[n];
        float a0 = dn * h_proj[((size_t)b0 * NN + n) * HH + h];  // +eye diagonal term
        float a1 = dn * h_proj[((size_t)b1 * NN + n) * HH + h];
        for (int w = 0; w < ROWW; ++w) {
            unsigned bits = row[w];
            while (bits) {
                int m = (w << 5) + __builtin_ctz(bits);
                bits &= bits - 1;
                float dm = dis[m];
                a0 += dm * h_proj[((size_t)b0 * NN + m) * HH + h];
                a1 += dm * h_proj[((size_t)b1 * NN + m) * HH + h];
            }
        }
        float z0 = fmaxf(a0 * dn + bg[h], 0.0f);
        float z1 = fmaxf(a1 * dn + bg[h], 0.0f);

        wz0 += z0 * x[((size_t)b0 * NN + n) * 2 + 1];
        wz1 += z1 * x[((size_t)b1 * NN + n) * 2 + 1];
        if (n == tg0) z_target[b0 * 64 + h] = z0;
        if (n == tg1) z_target[b1 * 64 + h] = z1;

        // softplus(z) = max(z,0) + log1p(exp(-|z|)); z>=0 after relu
        float sp0 = z0 + log1pf(expf(-z0));
        float sp1 = z1 + log1pf(expf(-z1));
        for (int o = 16; o; o >>= 1) { sp0 += __shfl_down(sp0, o); sp1 += __shfl_down(sp1, o); }
        if (lane == 0) { spw[wv] = sp0; spw[8 + wv] = sp1; }
        __syncthreads();
        if (t < 8) {
            int b = t;
            float spn = (b < 4) ? (spw[2 * b] + spw[2 * b + 1])
                                : (spw[8 + 2 * (b - 4)] + spw[8 + 2 * (b - 4) + 1]);
            spaccR += msk[(size_t)b * NN + n] * spn;
            nlaccR += x[((size_t)b * NN + n) * 2];
        }
    }
    atomicAdd(&wz_g[b0 * 64 + h], wz0);
    atomicAdd(&wz_g[b1 * 64 + h], wz1);
    if (t < 8) { atomicAdd(&sp_g[t], spaccR); atomicAdd(&nl_g[t], nlaccR); }
}

// ------------------------------------------------------------------ finalize
__global__ void k_final(const float* __restrict__ wz_g, const float* __restrict__ z_target,
                        const float* __restrict__ sp_g, const float* __restrict__ nl_g,
                        const float* __restrict__ W_d, const float* __restrict__ b_d,
                        const float* __restrict__ W_a, const float* __restrict__ b_a,
                        const float* __restrict__ prereq_w, float* __restrict__ out)
{
    int t = threadIdx.x;
    if (t < BB) {
        float nl = fmaxf(nl_g[t], 1.0f);
        float ab = 0.f, df = 0.f;
        for (int h = 0; h < HH; ++h) {
            ab += wz_g[t * 64 + h] * W_a[h];
            df += z_target[t * 64 + h] * W_d[h];
        }
        ab = ab / nl + b_a[0];
        df += b_d[0];
        float gap = ab - df + fabsf(prereq_w[0]) * (sp_g[t] * (1.0f / 64.0f));
        out[t]      = 1.0f / (1.0f + expf(-gap));
        out[BB + t] = gap;
    }
}

// ------------------------------------------------------------------- launcher
extern "C" void kernel_launch(void* const* d_in, const int* in_sizes, int n_in,
                              void* d_out, int out_size, void* d_ws, size_t ws_size,
                              hipStream_t stream)
{
    const float* x    = (const float*)d_in[0];
    const int*   ei   = (const int*)d_in[1];
    const int*   tgt  = (const int*)d_in[2];
    const float* msk  = (const float*)d_in[3];
    const float* nemb = (const float*)d_in[4];
    const float* Wsp  = (const float*)d_in[5];
    const float* bsp  = (const float*)d_in[6];
    const float* Wg   = (const float*)d_in[7];
    const float* bgp  = (const float*)d_in[8];
    const float* Wd   = (const float*)d_in[9];
    const float* bd   = (const float*)d_in[10];
    const float* Wa   = (const float*)d_in[11];
    const float* ba   = (const float*)d_in[12];
    const float* pw   = (const float*)d_in[13];
    float* out = (float*)d_out;
    int E = in_sizes[1] / 2;

    char* ws = (char*)d_ws;
    size_t off = 0;
    unsigned* bitmask = (unsigned*)(ws + off); off += (size_t)NN * ROWW * 4;  // 8 MB
    float* dis   = (float*)(ws + off);         off += (size_t)NN * 4;          // 32 KB
    float* hproj = (float*)(ws + off);         off += (size_t)BB * NN * HH * 4;// 16 MB
    float* wz_g  = (float*)(ws + off);         off += BB * HH * 4;
    float* ztgt  = (float*)(ws + off);         off += BB * HH * 4;
    float* sp_g  = (float*)(ws + off);         off += BB * 4;
    float* nl_g  = (float*)(ws + off);         off += BB * 4;

    hipMemsetAsync(bitmask, 0, (size_t)NN * ROWW * 4, stream);
    hipMemsetAsync(wz_g, 0, (2 * BB * HH + 2 * BB) * 4, stream);

    k_scatter<<<(E + 255) / 256, 256, 0, stream>>>(ei, bitmask, E);
    k_degree<<<(NN * 32) / 256, 256, 0, stream>>>(bitmask, dis);
    k_hproj<<<(BB * NN) / 128, 256, 0, stream>>>(x, nemb, Wsp, bsp, Wg, hproj);
    k_spmm<<<NN / 32, 256, 0, stream>>>(bitmask, dis, hproj, x, bgp, msk, tgt,
                                        wz_g, sp_g, nl_g, ztgt);
    k_final<<<1, 64, 0, stream>>>(wz_g, ztgt, sp_g, nl_g, Wd, bd, Wa, ba, pw, out);
}